// SecondOrderAgg_59992103190812
// MI455X (gfx1250) — compile-verified
//
#include <hip/hip_runtime.h>
#include <hip/hip_bf16.h>

typedef float v2f __attribute__((ext_vector_type(2)));
typedef float v8f __attribute__((ext_vector_type(8)));

#define D_DIM 768
#define N_SPATIAL 576
#define B_DIM 64
#define O_DIM 512
#define NCHUNK 4                 // split spatial dim into 4 chunks for parallelism
#define ROWS_PER_CHUNK (N_SPATIAL / NCHUNK)   // 144

// ---------------------------------------------------------------------------
// Kernel 1: streaming partial column statistics (bandwidth-bound phase).
// grid = (3*64, NCHUNK), block = 192 threads, each thread owns 4 columns
// (float4) -> 768 columns per block, fully coalesced 3 KB rows.
// Writes partial sum / sumsq per (chunk, feature, batch, d) to workspace.
// ---------------------------------------------------------------------------
__global__ __launch_bounds__(192)
void stats_partial_kernel(const float* __restrict__ f0,
                          const float* __restrict__ f1,
                          const float* __restrict__ f2,
                          float* __restrict__ sums,
                          float* __restrict__ sumsqs) {
    const int ib = blockIdx.x;          // i*64 + b, 0..191
    const int i  = ib / B_DIM;
    const int b  = ib % B_DIM;
    const int nc = blockIdx.y;          // 0..NCHUNK-1
    const int d0 = threadIdx.x * 4;     // 0..764

    const float* f = (i == 0) ? f0 : ((i == 1) ? f1 : f2);
    // skip the CLS token (row 0 of the 577 rows)
    const float* base = f + (size_t)b * 577 * D_DIM + D_DIM;

    float s0 = 0.f, s1 = 0.f, s2 = 0.f, s3 = 0.f;
    float q0 = 0.f, q1 = 0.f, q2 = 0.f, q3 = 0.f;

    const int n_begin = nc * ROWS_PER_CHUNK;
    const int n_end   = n_begin + ROWS_PER_CHUNK;
    for (int n = n_begin; n < n_end; ++n) {
        const float4 v = *(const float4*)(base + (size_t)n * D_DIM + d0);
        s0 += v.x; s1 += v.y; s2 += v.z; s3 += v.w;
        q0 += v.x * v.x; q1 += v.y * v.y; q2 += v.z * v.z; q3 += v.w * v.w;
    }

    const size_t off = ((size_t)(nc * 192 + ib)) * D_DIM + d0;
    float4 sv; sv.x = s0; sv.y = s1; sv.z = s2; sv.w = s3;
    float4 qv; qv.x = q0; qv.y = q1; qv.z = q2; qv.w = q3;
    *(float4*)(sums   + off) = sv;
    *(float4*)(sumsqs + off) = qv;
}

// ---------------------------------------------------------------------------
// Kernel 2: combine partials -> second-order vector, LayerNorm it, scale by
// alpha[i], accumulate onto cls_token. One block per batch row (64 blocks,
// 256 threads, each thread owns 3 of the 768 columns).
// ---------------------------------------------------------------------------
__global__ __launch_bounds__(256)
void aggregate_ln_kernel(const float* __restrict__ sums,
                         const float* __restrict__ sumsqs,
                         const float* __restrict__ cls,
                         const float* __restrict__ alpha,
                         const float* __restrict__ gamma,
                         const float* __restrict__ beta,
                         float* __restrict__ agg) {
    const int b = blockIdx.x;
    const int t = threadIdx.x;   // 0..255
    __shared__ float red[256];

    float acc[3];
    #pragma unroll
    for (int j = 0; j < 3; ++j) acc[j] = cls[(size_t)b * D_DIM + j * 256 + t];

    const float invN  = 1.0f / (float)N_SPATIAL;
    const float invN1 = 1.0f / (float)(N_SPATIAL - 1);

    for (int i = 0; i < 3; ++i) {
        float so[3];
        #pragma unroll
        for (int j = 0; j < 3; ++j) {
            const int d = j * 256 + t;
            float s = 0.f, q = 0.f;
            #pragma unroll
            for (int nc = 0; nc < NCHUNK; ++nc) {
                const size_t off = ((size_t)(nc * 192 + i * B_DIM + b)) * D_DIM + d;
                s += sums[off];
                q += sumsqs[off];
            }
            const float mean = s * invN;
            const float diag = (q - s * s * invN) * invN1;
            so[j] = mean + 0.1f * diag;
        }

        // Reduce mean over D
        float ls = so[0] + so[1] + so[2];
        red[t] = ls; __syncthreads();
        for (int w = 128; w > 0; w >>= 1) {
            if (t < w) red[t] += red[t + w];
            __syncthreads();
        }
        const float mu = red[0] * (1.0f / (float)D_DIM);
        __syncthreads();

        // Reduce E[x^2] over D
        float lq = so[0] * so[0] + so[1] * so[1] + so[2] * so[2];
        red[t] = lq; __syncthreads();
        for (int w = 128; w > 0; w >>= 1) {
            if (t < w) red[t] += red[t + w];
            __syncthreads();
        }
        const float var = red[0] * (1.0f / (float)D_DIM) - mu * mu;
        __syncthreads();

        const float rstd = rsqrtf(var + 1e-5f);
        const float al = alpha[i];
        #pragma unroll
        for (int j = 0; j < 3; ++j) {
            const int d = j * 256 + t;
            const float nrm = (so[j] - mu) * rstd * gamma[i * D_DIM + d] + beta[i * D_DIM + d];
            acc[j] += al * nrm;
        }
    }

    #pragma unroll
    for (int j = 0; j < 3; ++j) agg[(size_t)b * D_DIM + j * 256 + t] = acc[j];
}

// ---------------------------------------------------------------------------
// Kernels 3/4: fp32 WMMA GEMM, C = A(MxK) * B(KxN) + bias, optional exact GELU.
// One wave per 16x16 output tile, V_WMMA_F32_16X16X4_F32 along K in steps of 4.
// Fragment layout per CDNA5 ISA 7.12.2:
//   A 16x4 : lane l<16 -> row M=l, VGPR{0,1}=K{k0,k0+1}; lanes 16-31 -> K{k0+2,k0+3}
//   B 4x16 : N = lane%16 striped across lanes; VGPR{0,1} = K rows (k0[,+2],k0+1[,+3])
//   C/D    : VGPR r -> M = r + 8*(lane>=16), N = lane%16
// Launched with exact wave counts so EXEC is all-1s at every WMMA.
// ---------------------------------------------------------------------------
template <int GELU>
__global__ __launch_bounds__(256)
void gemm_wmma_f32_kernel(const float* __restrict__ A,
                          const float* __restrict__ B,
                          const float* __restrict__ bias,
                          float* __restrict__ C,
                          int M, int N, int K) {
    const int lane = threadIdx.x & 31;
    const int wave = threadIdx.x >> 5;
    const int gw   = blockIdx.x * (blockDim.x >> 5) + wave;
    const int ntiles = N >> 4;
    const int tm = gw / ntiles;
    const int tn = gw % ntiles;

    const int half = lane >> 4;      // 0 or 1
    const int l    = lane & 15;

    const float* Arow = A + (size_t)(tm * 16 + l) * K;
    const float* Bcol = B + tn * 16 + l;

    v8f c = {};
    for (int k = 0; k < K; k += 4) {
        v2f a, bf;
        a.x  = Arow[k + 2 * half];
        a.y  = Arow[k + 2 * half + 1];
        bf.x = Bcol[(size_t)(k + 2 * half) * N];
        bf.y = Bcol[(size_t)(k + 2 * half + 1) * N];
        c = __builtin_amdgcn_wmma_f32_16x16x4_f32(
                /*neg_a=*/false, a, /*neg_b=*/false, bf,
                /*c_mod=*/(short)0, c, /*reuse_a=*/false, /*reuse_b=*/false);
    }

    const int n = tn * 16 + l;
    const float bn = bias[n];
    #pragma unroll
    for (int r = 0; r < 8; ++r) {
        const int m = tm * 16 + r + 8 * half;
        float v = c[r] + bn;
        if (GELU) v = 0.5f * v * (1.0f + erff(v * 0.70710678118654752f));
        C[(size_t)m * N + n] = v;
    }
}

// ---------------------------------------------------------------------------
// Host-side orchestration.
// ---------------------------------------------------------------------------
extern "C" void kernel_launch(void* const* d_in, const int* in_sizes, int n_in,
                              void* d_out, int out_size, void* d_ws, size_t ws_size,
                              hipStream_t stream) {
    (void)in_sizes; (void)n_in; (void)out_size; (void)ws_size;

    const float* f0    = (const float*)d_in[0];
    const float* f1    = (const float*)d_in[1];
    const float* f2    = (const float*)d_in[2];
    const float* cls   = (const float*)d_in[3];
    const float* alpha = (const float*)d_in[4];
    const float* gamma = (const float*)d_in[5];
    const float* beta  = (const float*)d_in[6];
    const float* W1    = (const float*)d_in[7];
    const float* b1    = (const float*)d_in[8];
    const float* W2    = (const float*)d_in[9];
    const float* b2    = (const float*)d_in[10];
    float* out = (float*)d_out;

    // Workspace layout (floats):
    //   [0)                      partial sums   : NCHUNK*192*768 = 589824
    //   [589824)                 partial sumsqs : 589824
    //   [1179648)                aggregated     : 64*768 = 49152
    //   [1228800)                hidden h       : 64*768 = 49152
    float* ws      = (float*)d_ws;
    float* sums    = ws;
    float* sumsqs  = ws + (size_t)NCHUNK * 192 * D_DIM;
    float* agg     = sumsqs + (size_t)NCHUNK * 192 * D_DIM;
    float* hbuf    = agg + (size_t)B_DIM * D_DIM;

    // Phase 1: streaming reduction (bandwidth-bound, ~340 MB read)
    stats_partial_kernel<<<dim3(3 * B_DIM, NCHUNK), 192, 0, stream>>>(
        f0, f1, f2, sums, sumsqs);

    // Phase 2: combine + layernorm + alpha-weighted aggregation with cls token
    aggregate_ln_kernel<<<B_DIM, 256, 0, stream>>>(
        sums, sumsqs, cls, alpha, gamma, beta, agg);

    // Phase 3: h = GELU(agg @ W1 + b1)   -> 4x48 = 192 tiles = 24 blocks of 8 waves
    gemm_wmma_f32_kernel<1><<<(B_DIM / 16) * (D_DIM / 16) / 8, 256, 0, stream>>>(
        agg, W1, b1, hbuf, B_DIM, D_DIM, D_DIM);

    // Phase 4: out = h @ W2 + b2         -> 4x32 = 128 tiles = 16 blocks of 8 waves
    gemm_wmma_f32_kernel<0><<<(B_DIM / 16) * (O_DIM / 16) / 8, 256, 0, stream>>>(
        hbuf, W2, b2, out, B_DIM, O_DIM, D_DIM);
}